// SSIMLoss_19825569038716
// MI455X (gfx1250) — compile-verified
//
#include <hip/hip_runtime.h>
#include <hip/hip_bf16.h>

typedef float v2f __attribute__((ext_vector_type(2)));
typedef float v8f __attribute__((ext_vector_type(8)));

#define IND   384
#define OUTD  378
#define TILES 24
#define SSIM_C1 (0.01f * 0.01f)
#define SSIM_C2 (0.03f * 0.03f)
#define COVN    (49.0f / 48.0f)

__device__ __forceinline__ float bandw(int d, float v) {
    return (d >= 0 && d <= 6) ? v : 0.0f;
}

__global__ __launch_bounds__(32) void ssim_init_kernel(float* acc) {
    if (threadIdx.x == 0) acc[0] = 0.0f;
}

__global__ __launch_bounds__(32) void ssim_final_kernel(const float* acc, float* out) {
    if (threadIdx.x == 0) {
        const float cnt = (float)(64 * OUTD * OUTD);
        out[0] = 1.0f - acc[0] / cnt;
    }
}

// One wave (32 threads) computes one 16x16 output tile of the SSIM map.
// Both 7-tap box-filter passes are banded-matrix matmuls on V_WMMA_F32_16X16X4_F32.
__global__ __launch_bounds__(32) void ssim_tile_kernel(
    const float* __restrict__ X, const float* __restrict__ Y,
    float* __restrict__ acc)
{
    // Per-channel D1 (vertical-filtered 16x22) staging; stride 33 floats for
    // conflict-free column reads in stage 2.
    __shared__ float lds[5][16 * 33];

    const int l  = threadIdx.x;
    const int m  = l & 15;          // A-operand M row / B-operand N column / C-layout N
    const int hk = (l >> 4) * 2;    // K offset for lanes 16..31 (per WMMA f32 layout)

    const int tx = blockIdx.x, ty = blockIdx.y, img = blockIdx.z;
    const int row0 = ty * 16, col0 = tx * 16;

    const float* Xb = X + (size_t)img * IND * IND;
    const float* Yb = Y + (size_t)img * IND * IND;

    // ---- Load 24-row x 32-col clamped patch in B-operand register layout ----
    float xv[6][2][2], yv[6][2][2];   // [k-chunk][vgpr][n-half]
    #pragma unroll
    for (int kc = 0; kc < 6; ++kc) {
        #pragma unroll
        for (int vg = 0; vg < 2; ++vg) {
            int r = row0 + 4 * kc + hk + vg;
            r = r > (IND - 1) ? (IND - 1) : r;
            #pragma unroll
            for (int nh = 0; nh < 2; ++nh) {
                int c = col0 + nh * 16 + m;
                c = c > (IND - 1) ? (IND - 1) : c;
                const size_t idx = (size_t)r * IND + c;
                xv[kc][vg][nh] = Xb[idx];
                yv[kc][vg][nh] = Yb[idx];
            }
        }
    }

    const float inv49 = 1.0f / 49.0f;

    // ---- Stage 1: vertical band matmul  D1 = Wv(16x24, 1/49) x In(24x{16,16}) ----
    #pragma unroll
    for (int ch = 0; ch < 5; ++ch) {
        v8f d1a = {};
        v8f d1b = {};
        #pragma unroll
        for (int kc = 0; kc < 6; ++kc) {
            const int k0 = 4 * kc + hk;
            v2f A;
            A.x = bandw(k0 - m, inv49);
            A.y = bandw(k0 + 1 - m, inv49);
            #pragma unroll
            for (int nh = 0; nh < 2; ++nh) {
                const float x0 = xv[kc][0][nh], x1 = xv[kc][1][nh];
                const float y0 = yv[kc][0][nh], y1 = yv[kc][1][nh];
                float b0, b1;
                if      (ch == 0) { b0 = x0;      b1 = x1;      }
                else if (ch == 1) { b0 = y0;      b1 = y1;      }
                else if (ch == 2) { b0 = x0 * x0; b1 = x1 * x1; }
                else if (ch == 3) { b0 = y0 * y0; b1 = y1 * y1; }
                else              { b0 = x0 * y0; b1 = x1 * y1; }
                v2f B; B.x = b0; B.y = b1;
                if (nh == 0)
                    d1a = __builtin_amdgcn_wmma_f32_16x16x4_f32(
                        false, A, false, B, (short)0, d1a, false, false);
                else
                    d1b = __builtin_amdgcn_wmma_f32_16x16x4_f32(
                        false, A, false, B, (short)0, d1b, false, false);
            }
        }
        // C layout -> LDS (row-major, stride 33)
        #pragma unroll
        for (int v = 0; v < 8; ++v) {
            const int rr = v + 8 * (l >> 4);
            lds[ch][rr * 33 + m]      = d1a[v];
            lds[ch][rr * 33 + 16 + m] = d1b[v];
        }
    }

    __syncthreads();

    // ---- Stage 2: horizontal band matmul  Out^T = Wh(16x24, 1.0) x D1^T ----
    v8f outc[5];
    #pragma unroll
    for (int ch = 0; ch < 5; ++ch) {
        v8f d2 = {};
        #pragma unroll
        for (int jc = 0; jc < 6; ++jc) {
            const int j0 = 4 * jc + hk;
            v2f A;
            A.x = bandw(j0 - m, 1.0f);
            A.y = bandw(j0 + 1 - m, 1.0f);
            v2f B;                       // B(j, r) = D1(r=m, col=j)
            B.x = lds[ch][m * 33 + j0];
            B.y = lds[ch][m * 33 + j0 + 1];
            d2 = __builtin_amdgcn_wmma_f32_16x16x4_f32(
                false, A, false, B, (short)0, d2, false, false);
        }
        outc[ch] = d2;
    }

    // ---- SSIM elementwise math + masked reduction ----
    // Out^T C layout: element v of lane l -> out row (row0 + m), col (col0 + v + 8*(l>>4))
    float a = 0.0f;
    const int rg = row0 + m;
    #pragma unroll
    for (int v = 0; v < 8; ++v) {
        const int cg = col0 + v + 8 * (l >> 4);
        const float ux  = outc[0][v];
        const float uy  = outc[1][v];
        const float uxx = outc[2][v];
        const float uyy = outc[3][v];
        const float uxy = outc[4][v];
        const float vx  = COVN * (uxx - ux * ux);
        const float vy  = COVN * (uyy - uy * uy);
        const float vxy = COVN * (uxy - ux * uy);
        const float A1 = 2.0f * ux * uy + SSIM_C1;
        const float A2 = 2.0f * vxy + SSIM_C2;
        const float B1 = ux * ux + uy * uy + SSIM_C1;
        const float B2 = vx + vy + SSIM_C2;
        const float S  = (A1 * A2) / (B1 * B2);
        if (rg < OUTD && cg < OUTD) a += S;
    }

    // wave32 reduction
    #pragma unroll
    for (int off = 16; off > 0; off >>= 1)
        a += __shfl_xor(a, off, 32);

    if (l == 0) atomicAdd(acc, a);
}

extern "C" void kernel_launch(void* const* d_in, const int* in_sizes, int n_in,
                              void* d_out, int out_size, void* d_ws, size_t ws_size,
                              hipStream_t stream) {
    (void)in_sizes; (void)n_in; (void)out_size; (void)ws_size;
    const float* X = (const float*)d_in[0];
    const float* Y = (const float*)d_in[1];
    // d_in[2] is the 7x7 ones/49 weight; constant, folded into Wv.
    float* acc = (float*)d_ws;
    float* out = (float*)d_out;

    ssim_init_kernel<<<1, 32, 0, stream>>>(acc);
    ssim_tile_kernel<<<dim3(TILES, TILES, 64), 32, 0, stream>>>(X, Y, acc);
    ssim_final_kernel<<<1, 32, 0, stream>>>(acc, out);
}